// PropagationLayer_70729521430968
// MI455X (gfx1250) — compile-verified
//
#include <hip/hip_runtime.h>

#define B_    8
#define V_    8192
#define H_    128
#define K_    128
#define NNZ_  57344
#define N_    (B_*V_)    // 65536 rows
#define H2_   256
#define H3_   384
#define EPS_BN 1e-5f
#define SCALE_ 1.0f

typedef float v2f __attribute__((ext_vector_type(2)));
typedef float v8f __attribute__((ext_vector_type(8)));

// D(16x16 f32) = A(16x4 f32) * B(4x16 f32) + C   -> v_wmma_f32_16x16x4_f32
__device__ __forceinline__ v8f wmma4(v2f a, v2f b, v8f c) {
  return __builtin_amdgcn_wmma_f32_16x16x4_f32(
      /*neg_a=*/false, a, /*neg_b=*/false, b,
      /*c_mod=*/(short)0, c, /*reuse_a=*/false, /*reuse_b=*/false);
}

__device__ __forceinline__ v2f ldv2(const float* p) { return *(const v2f*)p; }

// ---------------------------------------------------------------- utilities
__global__ void k_zero(float* __restrict__ p, long long n) {
  long long i = (long long)blockIdx.x * blockDim.x + threadIdx.x;
  long long stride = (long long)gridDim.x * blockDim.x;
  for (; i < n; i += stride) p[i] = 0.0f;
}

// ---------------------------------------------------------------- spectral
// h_spec[b,k,h] = sum_v Uinv_rows[v,k] * h[b,v,h]  (A^T*B), split-K over v.
// 32x32 macro-tile per wave: 4 WMMA per 8 b32 loads.
// grid: x = 16 macro-tiles (4x4), y = b, z = slice ; block = 32
__global__ void k_spec(const float* __restrict__ eigs, const float* __restrict__ h,
                       float* __restrict__ hm) {
  const int lane = threadIdx.x, half = lane >> 4, l16 = lane & 15;
  const int kt = (blockIdx.x >> 2) * 32, ht = (blockIdx.x & 3) * 32;
  const int b = blockIdx.y;
  const int nslice = gridDim.z;
  const float* Ui = eigs + (size_t)b*(1+2*V_)*K_ + (size_t)(1+V_)*K_; // [V][K]
  const float* hb = h + (size_t)b*V_*H_;
  v8f c00 = {0.f,0.f,0.f,0.f,0.f,0.f,0.f,0.f};
  v8f c01 = c00, c10 = c00, c11 = c00;
  const int vchunk = V_ / nslice;
  const int v0s = blockIdx.z * vchunk;
  for (int v0 = v0s; v0 < v0s + vchunk; v0 += 4) {
    size_t r0 = (size_t)(v0 + 2*half)*K_;
    size_t r1 = (size_t)(v0 + 2*half + 1)*K_;
    size_t s0 = (size_t)(v0 + 2*half)*H_;
    size_t s1 = (size_t)(v0 + 2*half + 1)*H_;
    v2f a0, a1, b0, b1;
    a0.x = Ui[r0 + kt + l16];        a0.y = Ui[r1 + kt + l16];
    a1.x = Ui[r0 + kt + 16 + l16];   a1.y = Ui[r1 + kt + 16 + l16];
    b0.x = hb[s0 + ht + l16];        b0.y = hb[s1 + ht + l16];
    b1.x = hb[s0 + ht + 16 + l16];   b1.y = hb[s1 + ht + 16 + l16];
    c00 = wmma4(a0, b0, c00);
    c01 = wmma4(a0, b1, c01);
    c10 = wmma4(a1, b0, c10);
    c11 = wmma4(a1, b1, c11);
  }
  float* out = hm + (size_t)b*K_*H_;
#pragma unroll
  for (int r = 0; r < 8; ++r) {
    size_t row0 = (size_t)(kt + r + 8*half)*H_;
    size_t row1 = (size_t)(kt + 16 + r + 8*half)*H_;
    atomicAdd(&out[row0 + ht + l16],      c00[r]);
    atomicAdd(&out[row0 + ht + 16 + l16], c01[r]);
    atomicAdd(&out[row1 + ht + l16],      c10[r]);
    atomicAdd(&out[row1 + ht + 16 + l16], c11[r]);
  }
}

// hm[b,k,h] *= band(k,h) * prop(k,h)
__global__ void k_modulate(const float* __restrict__ eigs, const float* __restrict__ pt,
                           const float* __restrict__ bem, const float* __restrict__ bes,
                           float* __restrict__ hm) {
  int i = blockIdx.x * blockDim.x + threadIdx.x;
  if (i >= B_*K_*H_) return;
  int hcol = i % H_;
  int k    = (i / H_) % K_;
  int b    = i / (H_*K_);
  float ev = eigs[(size_t)b*(1+2*V_)*K_ + k];           // row 0 = eigenvalues
  float t  = SCALE_ * fmaxf(pt[hcol], 1e-6f);
  float m  = fminf(fmaxf(bem[hcol], 1e-6f), 0.3f);
  float s  = fmaxf(bes[hcol], 0.05f);
  float d  = m - ev;
  hm[i] *= __expf(-(d*d) / (2.0f*s*s)) * __expf(-ev * t);
}

// h_prop[b,v,h] = sum_k U[v,k] * hm[b,k,h]
// Block = 256 (8 waves); each wave owns one 16-row band and all 8 h-tiles.
// hm staged through LDS in 32KB chunks; A loaded once (b64), 8 WMMA per step.
// grid: x = V/(16*8) = 64, z = b
__global__ void k_prop(const float* __restrict__ eigs, const float* __restrict__ hm,
                       float* __restrict__ hprop) {
  __shared__ float smem[64 * H_];                       // 32 KB
  const int tid = threadIdx.x;
  const int wave = tid >> 5, lane = tid & 31, half = lane >> 4, l16 = lane & 15;
  const int b = blockIdx.z;
  const int v0 = (blockIdx.x * 8 + wave) * 16;
  const float* U = eigs + (size_t)b*(1+2*V_)*K_ + K_;   // rows 1..V
  const float* M = hm + (size_t)b*K_*H_;
  v8f acc[8];
#pragma unroll
  for (int t = 0; t < 8; ++t) acc[t] = (v8f){0.f,0.f,0.f,0.f,0.f,0.f,0.f,0.f};

  for (int k0 = 0; k0 < K_; k0 += 64) {
    __syncthreads();
    for (int i = tid * 4; i < 64 * H_; i += 256 * 4)    // 8 x b128 per thread
      *(float4*)(smem + i) = *(const float4*)(M + (size_t)k0*H_ + i);
    __syncthreads();
    for (int kk = 0; kk < 64; kk += 4) {
      v2f a = ldv2(U + (size_t)(v0 + l16)*K_ + k0 + kk + 2*half);
      int sr0 = (kk + 2*half) * H_, sr1 = (kk + 2*half + 1) * H_;
#pragma unroll
      for (int t = 0; t < 8; ++t) {
        v2f bb;
        bb.x = smem[sr0 + t*16 + l16];
        bb.y = smem[sr1 + t*16 + l16];
        acc[t] = wmma4(a, bb, acc[t]);
      }
    }
  }
  float* out = hprop + (size_t)b*V_*H_;
#pragma unroll
  for (int r = 0; r < 8; ++r) {
    size_t row = (size_t)(v0 + r + 8*half)*H_;
#pragma unroll
    for (int t = 0; t < 8; ++t)
      out[row + t*16 + l16] = acc[t][r];
  }
}

// ---------------------------------------------------------------- scatter
__global__ void k_scatter(const float* __restrict__ h, const float* __restrict__ gvals,
                          const int* __restrict__ gidx,
                          float* __restrict__ gradx, float* __restrict__ grady) {
  int t = blockIdx.x * blockDim.x + threadIdx.x;
  if (t >= B_*NNZ_) return;
  int b = t / NNZ_, e = t % NNZ_;
  const int*   gi = gidx  + (size_t)b*2*NNZ_;
  const float* gv = gvals + (size_t)b*2*NNZ_;
  int row = gi[e], col = gi[NNZ_ + e];
  float g0 = gv[e], g1 = gv[NNZ_ + e];
  const float4* hc = (const float4*)(h + ((size_t)b*V_ + col)*H_);
  __builtin_prefetch(hc, 0, 0);                         // global_prefetch_b8
  float* gx = gradx + ((size_t)b*V_ + row)*H_;
  float* gy = grady + ((size_t)b*V_ + row)*H_;
  for (int q = 0; q < H_/4; ++q) {
    float4 hv = hc[q];
    atomicAdd(&gx[q*4 + 0], g0 * hv.x);
    atomicAdd(&gx[q*4 + 1], g0 * hv.y);
    atomicAdd(&gx[q*4 + 2], g0 * hv.z);
    atomicAdd(&gx[q*4 + 3], g0 * hv.w);
    atomicAdd(&gy[q*4 + 0], g1 * hv.x);
    atomicAdd(&gy[q*4 + 1], g1 * hv.y);
    atomicAdd(&gy[q*4 + 2], g1 * hv.z);
    atomicAdd(&gy[q*4 + 3], g1 * hv.w);
  }
}

// ---------------------------------------------------------------- grad path
// Fused 4-matrix GEMM + cross identity. 16x32 macro-tile: 8 WMMA per 6 b64 loads.
// grid: x = V/16, y = H/32, z = b ; block = 32
__global__ void k_grad(const float* __restrict__ gradx, const float* __restrict__ grady,
                       const float* __restrict__ Wg, const float* __restrict__ Wg2,
                       const float* __restrict__ gb, const float* __restrict__ vn,
                       float* __restrict__ hgrad) {
  const int lane = threadIdx.x, half = lane >> 4, l16 = lane & 15;
  const int v0 = blockIdx.x * 16, h0 = blockIdx.y * 32, b = blockIdx.z;
  const float* GX = gradx + (size_t)b*V_*H_;
  const float* GY = grady + (size_t)b*V_*H_;
  v8f cxg0 = {0.f,0.f,0.f,0.f,0.f,0.f,0.f,0.f};
  v8f cxg1 = cxg0, cyg0 = cxg0, cyg1 = cxg0;
  v8f cxp0 = cxg0, cxp1 = cxg0, cyp0 = cxg0, cyp1 = cxg0;
  for (int j = 0; j < H_; j += 4) {
    size_t ra  = (size_t)(v0 + l16)*H_ + j + 2*half;
    size_t rb0 = (size_t)(h0 + l16)*H_ + j + 2*half;
    size_t rb1 = (size_t)(h0 + 16 + l16)*H_ + j + 2*half;
    v2f ax = ldv2(GX + ra), ay = ldv2(GY + ra);
    ax.x = fminf(fmaxf(ax.x, -1.f), 1.f);  ax.y = fminf(fmaxf(ax.y, -1.f), 1.f);
    ay.x = fminf(fmaxf(ay.x, -1.f), 1.f);  ay.y = fminf(fmaxf(ay.y, -1.f), 1.f);
    v2f bg0  = ldv2(Wg  + rb0), bg1  = ldv2(Wg  + rb1);
    v2f bp0  = ldv2(Wg2 + rb0), bp1  = ldv2(Wg2 + rb1);
    cxg0 = wmma4(ax, bg0, cxg0);  cxg1 = wmma4(ax, bg1, cxg1);
    cyg0 = wmma4(ay, bg0, cyg0);  cyg1 = wmma4(ay, bg1, cyg1);
    cxp0 = wmma4(ax, bp0, cxp0);  cxp1 = wmma4(ax, bp1, cxp1);
    cyp0 = wmma4(ay, bp0, cyp0);  cyp1 = wmma4(ay, bp1, cyp1);
  }
  float* out = hgrad + (size_t)b*V_*H_;
#pragma unroll
  for (int r = 0; r < 8; ++r) {
    int v = v0 + r + 8*half;
    const float* g = gb + ((size_t)b*V_ + v)*6;
    const float* n = vn + ((size_t)b*V_ + v)*3;
    float X0=g[0], X1=g[1], X2=g[2], Y0=g[3], Y1=g[4], Y2=g[5];
    float cx = X1*Y2 - X2*Y1, cy = X2*Y0 - X0*Y2, cz = X0*Y1 - X1*Y0;
    float s  = cx*n[0] + cy*n[1] + cz*n[2];
    size_t row = (size_t)v*H_;
    out[row + h0 + l16]      = (cxg0[r]*cyp0[r] - cyg0[r]*cxp0[r]) * s;
    out[row + h0 + 16 + l16] = (cxg1[r]*cyp1[r] - cyg1[r]*cxp1[r]) * s;
  }
}

// ---------------------------------------------------------------- MLP
// y1 = concat(hf,hprop,hgrad) @ W1.T + b1.  32x32 macro-tile: 4 WMMA / 4 b64.
// grid: x = N/32, y = H2/32 ; block = 32
__global__ void k_mlp1(const float* __restrict__ hf, const float* __restrict__ hprop,
                       const float* __restrict__ hgrad, const float* __restrict__ W1,
                       const float* __restrict__ b1, float* __restrict__ y1) {
  const int lane = threadIdx.x, half = lane >> 4, l16 = lane & 15;
  const int n0 = blockIdx.x * 32, j0 = blockIdx.y * 32;
  v8f c00 = {0.f,0.f,0.f,0.f,0.f,0.f,0.f,0.f};
  v8f c01 = c00, c10 = c00, c11 = c00;
  for (int cc = 0; cc < H3_; cc += 4) {
    const float* src = (cc < H_) ? hf : ((cc < 2*H_) ? hprop : hgrad);
    int ci = cc & (H_ - 1);
    v2f a0 = ldv2(src + (size_t)(n0 + l16)*H_ + ci + 2*half);
    v2f a1 = ldv2(src + (size_t)(n0 + 16 + l16)*H_ + ci + 2*half);
    v2f b0 = ldv2(W1 + (size_t)(j0 + l16)*H3_ + cc + 2*half);
    v2f b1v = ldv2(W1 + (size_t)(j0 + 16 + l16)*H3_ + cc + 2*half);
    c00 = wmma4(a0, b0, c00);   c01 = wmma4(a0, b1v, c01);
    c10 = wmma4(a1, b0, c10);   c11 = wmma4(a1, b1v, c11);
  }
  float bias0 = b1[j0 + l16], bias1 = b1[j0 + 16 + l16];
#pragma unroll
  for (int r = 0; r < 8; ++r) {
    size_t row0 = (size_t)(n0 + r + 8*half)*H2_;
    size_t row1 = (size_t)(n0 + 16 + r + 8*half)*H2_;
    y1[row0 + j0 + l16]      = c00[r] + bias0;
    y1[row0 + j0 + 16 + l16] = c01[r] + bias1;
    y1[row1 + j0 + l16]      = c10[r] + bias0;
    y1[row1 + j0 + 16 + l16] = c11[r] + bias1;
  }
}

// per-column sum / sumsq (256 row-slices, one partial per thread)
__global__ void k_colstats(const float* __restrict__ M, int nrows, int ncols,
                           float* __restrict__ stats) {
  int t = blockIdx.x * blockDim.x + threadIdx.x;
  if (t >= ncols * 256) return;
  int col = t % ncols, slice = t / ncols;
  float s = 0.f, s2 = 0.f;
  for (int r = slice; r < nrows; r += 256) {
    float v = M[(size_t)r*ncols + col];
    s += v; s2 += v*v;
  }
  atomicAdd(&stats[col], s);
  atomicAdd(&stats[ncols + col], s2);
}

__global__ void k_finstats(float* __restrict__ stats, int ncols, float nrows) {
  int c = blockIdx.x * blockDim.x + threadIdx.x;
  if (c >= ncols) return;
  float mean = stats[c] / nrows;
  float var  = stats[ncols + c] / nrows - mean*mean;
  stats[2*ncols + c] = mean;
  stats[3*ncols + c] = rsqrtf(var + EPS_BN);
}

// y2 = relu(bn1(y1)) @ W2.T + b2.  32x32 macro-tile.
// grid: x = N/32, y = H/32 ; block = 32
__global__ void k_mlp2(const float* __restrict__ y1, const float* __restrict__ st1,
                       const float* __restrict__ g1, const float* __restrict__ be1,
                       const float* __restrict__ W2, const float* __restrict__ b2,
                       float* __restrict__ y2) {
  const int lane = threadIdx.x, half = lane >> 4, l16 = lane & 15;
  const int n0 = blockIdx.x * 32, h0 = blockIdx.y * 32;
  const float* mean1 = st1 + 2*H2_;
  const float* inv1  = st1 + 3*H2_;
  v8f c00 = {0.f,0.f,0.f,0.f,0.f,0.f,0.f,0.f};
  v8f c01 = c00, c10 = c00, c11 = c00;
  for (int j = 0; j < H2_; j += 4) {
    int ja = j + 2*half;
    v2f m = ldv2(mean1 + ja), iv = ldv2(inv1 + ja);
    v2f gg = ldv2(g1 + ja),   bt = ldv2(be1 + ja);
    v2f r0 = ldv2(y1 + (size_t)(n0 + l16)*H2_ + ja);
    v2f r1 = ldv2(y1 + (size_t)(n0 + 16 + l16)*H2_ + ja);
    v2f a0, a1;
    a0.x = fmaxf(0.f, gg.x*(r0.x - m.x)*iv.x + bt.x);
    a0.y = fmaxf(0.f, gg.y*(r0.y - m.y)*iv.y + bt.y);
    a1.x = fmaxf(0.f, gg.x*(r1.x - m.x)*iv.x + bt.x);
    a1.y = fmaxf(0.f, gg.y*(r1.y - m.y)*iv.y + bt.y);
    v2f b0 = ldv2(W2 + (size_t)(h0 + l16)*H2_ + ja);
    v2f b1v = ldv2(W2 + (size_t)(h0 + 16 + l16)*H2_ + ja);
    c00 = wmma4(a0, b0, c00);   c01 = wmma4(a0, b1v, c01);
    c10 = wmma4(a1, b0, c10);   c11 = wmma4(a1, b1v, c11);
  }
  float bias0 = b2[h0 + l16], bias1 = b2[h0 + 16 + l16];
#pragma unroll
  for (int r = 0; r < 8; ++r) {
    size_t row0 = (size_t)(n0 + r + 8*half)*H_;
    size_t row1 = (size_t)(n0 + 16 + r + 8*half)*H_;
    y2[row0 + h0 + l16]      = c00[r] + bias0;
    y2[row0 + h0 + 16 + l16] = c01[r] + bias1;
    y2[row1 + h0 + l16]      = c10[r] + bias0;
    y2[row1 + h0 + 16 + l16] = c11[r] + bias1;
  }
}

__global__ void k_out(const float* __restrict__ hf, const float* __restrict__ y2,
                      const float* __restrict__ st2, const float* __restrict__ g2,
                      const float* __restrict__ be2, float* __restrict__ out) {
  int i = blockIdx.x * blockDim.x + threadIdx.x;   // float4 granules
  if (i >= N_*H_/4) return;
  int hcol = (i * 4) % H_;
  float4 y = ((const float4*)y2)[i];
  float4 hv = ((const float4*)hf)[i];
  float4 o;
  o.x = hv.x + g2[hcol+0]*(y.x - st2[2*H_+hcol+0])*st2[3*H_+hcol+0] + be2[hcol+0];
  o.y = hv.y + g2[hcol+1]*(y.y - st2[2*H_+hcol+1])*st2[3*H_+hcol+1] + be2[hcol+1];
  o.z = hv.z + g2[hcol+2]*(y.z - st2[2*H_+hcol+2])*st2[3*H_+hcol+2] + be2[hcol+2];
  o.w = hv.w + g2[hcol+3]*(y.w - st2[2*H_+hcol+3])*st2[3*H_+hcol+3] + be2[hcol+3];
  ((float4*)out)[i] = o;
}

// ---------------------------------------------------------------- launcher
extern "C" void kernel_launch(void* const* d_in, const int* in_sizes, int n_in,
                              void* d_out, int out_size, void* d_ws, size_t ws_size,
                              hipStream_t stream) {
  const float* h     = (const float*)d_in[0];
  const float* eigs  = (const float*)d_in[1];
  const float* gvals = (const float*)d_in[2];
  const int*   gidx  = (const int*)  d_in[3];
  const float* gb    = (const float*)d_in[4];
  const float* vn    = (const float*)d_in[5];
  const float* pt    = (const float*)d_in[6];
  const float* bem   = (const float*)d_in[7];
  const float* bes   = (const float*)d_in[8];
  const float* Wg    = (const float*)d_in[9];
  const float* Wg2   = (const float*)d_in[10];
  const float* W1    = (const float*)d_in[11];
  const float* b1    = (const float*)d_in[12];
  const float* g1    = (const float*)d_in[13];
  const float* be1   = (const float*)d_in[14];
  const float* W2    = (const float*)d_in[15];
  const float* b2    = (const float*)d_in[16];
  const float* g2    = (const float*)d_in[17];
  const float* be2   = (const float*)d_in[18];
  float* out = (float*)d_out;

  float* ws    = (float*)d_ws;
  float* hm    = ws;                              // B*K*H       = 131072
  float* hprop = hm    + (size_t)B_*K_*H_;        // N*H
  float* gradx = hprop + (size_t)N_*H_;           // N*H
  float* grady = gradx + (size_t)N_*H_;           // N*H
  float* hgrad = grady + (size_t)N_*H_;           // N*H
  float* y1    = hgrad + (size_t)N_*H_;           // N*H2
  float* y2    = y1    + (size_t)N_*H2_;          // N*H
  float* s1    = y2    + (size_t)N_*H_;           // 4*H2
  float* s2    = s1    + 4*H2_;                   // 4*H

  // zero the atomic accumulators (gradx|grady contiguous, hm, stats)
  k_zero<<<8192, 256, 0, stream>>>(gradx, (long long)2*N_*H_);
  k_zero<<<64,   256, 0, stream>>>(hm,    (long long)B_*K_*H_);
  k_zero<<<2,    256, 0, stream>>>(s1,    (long long)(4*H2_ + 4*H_));

  // spectral propagation
  k_spec<<<dim3(16, B_, 32), 32, 0, stream>>>(eigs, h, hm);
  k_modulate<<<(B_*K_*H_ + 255)/256, 256, 0, stream>>>(eigs, pt, bem, bes, hm);
  k_prop<<<dim3(V_/(16*8), 1, B_), 256, 0, stream>>>(eigs, hm, hprop);

  // sparse gradient accumulation + fused grad GEMMs/cross
  k_scatter<<<(B_*NNZ_ + 255)/256, 256, 0, stream>>>(h, gvals, gidx, gradx, grady);
  k_grad<<<dim3(V_/16, H_/32, B_), 32, 0, stream>>>(gradx, grady, Wg, Wg2, gb, vn, hgrad);

  // MLP with batch-norm
  k_mlp1<<<dim3(N_/32, H2_/32), 32, 0, stream>>>(h, hprop, hgrad, W1, b1, y1);
  k_colstats<<<(H2_*256)/256, 256, 0, stream>>>(y1, N_, H2_, s1);
  k_finstats<<<1, H2_, 0, stream>>>(s1, H2_, (float)N_);
  k_mlp2<<<dim3(N_/32, H_/32), 32, 0, stream>>>(y1, s1, g1, be1, W2, b2, y2);
  k_colstats<<<(H_*256)/256, 256, 0, stream>>>(y2, N_, H_, s2);
  k_finstats<<<1, H_, 0, stream>>>(s2, H_, (float)N_);
  k_out<<<(N_*H_/4 + 255)/256, 256, 0, stream>>>(h, y2, s2, g2, be2, out);
}